// LengthRegulator_65077344469577
// MI455X (gfx1250) — compile-verified
//
#include <hip/hip_runtime.h>
#include <hip/hip_bf16.h>

// ---------------------------------------------------------------------------
// FastSpeech duration predictor + length regulator for MI455X (gfx1250).
//
// Precision: duration path is round(exp(.)) -> cumsum -> searchsorted, which
// is discontinuous -> keep fp32 end-to-end using V_WMMA_F32_16X16X4_F32.
//
// Roofline: 2x 29 GFLOP fp32 GEMMs vs 226 MB output write (~10us @ 23.3TB/s).
// The limiting term is L2 weight traffic, so each block now computes a
// 32-token x 384-channel tile: every B (weight) fragment fetched from L2
// feeds two WMMAs, halving L2 traffic vs a 16-row tile.
// The activation slab is staged with GLOBAL_LOAD_ASYNC_TO_LDS_B128
// (ASYNCcnt path, no VGPR round-trip).
// ---------------------------------------------------------------------------

#define B_    32
#define T_    1024
#define D_    384
#define KTAPS 3
#define KDIM  (KTAPS * D_)   // 1152 = im2col K for the K=3 SAME conv
#define MROWS 32             // tokens per block
#define MAXF  4608
#define EPS_  1e-5f

typedef float v2f __attribute__((ext_vector_type(2)));
typedef float v8f __attribute__((ext_vector_type(8)));

__device__ __forceinline__ uint32_t lds_addr_of(const void* p) {
    // LDS aperture: flat addr low 32 bits == LDS byte offset (ISA 10.2).
    return (uint32_t)(uintptr_t)p;
}

// ---------------------------------------------------------------------------
// Fused conv1d(K=3, SAME) + bias + ReLU + LayerNorm.
// Block = 32 tokens x 384 channels. 8 waves, each: 2 row-tiles x 3 col-tiles.
// WMMA f32 16x16x4 layouts per ISA 7.12.2:
//   A 16x4 : lanes 0-15 -> {K=kk0, kk0+1}, lanes 16-31 -> {kk0+2, kk0+3}
//   B 4x16 : lanes 0-15 -> rows kk0/kk0+1 at col N=lane,
//            lanes 16-31 -> rows kk0+2/kk0+3 at col N=lane-16
//   C/D    : VGPR j, lane l<16 -> (M=j, N=l); lane l>=16 -> (M=j+8, N=l-16)
// ---------------------------------------------------------------------------
__global__ __launch_bounds__(256) void conv_ln_kernel(
    const float* __restrict__ in,    // [B, T, D]
    const float* __restrict__ W,     // [K, D, D]  (tap, cin, cout)
    const float* __restrict__ bias,  // [D]
    const float* __restrict__ gam,   // [D]
    const float* __restrict__ bet,   // [D]
    float* __restrict__ out)         // [B, T, D]
{
    // smem serves two phases: (1) 34x384 halo slab for the K loop,
    // (2) 32x384 post-ReLU tile for the LayerNorm pass (slab is dead then).
    __shared__ float smem[(MROWS + 2) * D_];   // 52224 B
    __shared__ float partS[MROWS * 8];
    __shared__ float partQ[MROWS * 8];
    __shared__ float rowMean[MROWS];
    __shared__ float rowRstd[MROWS];

    float* xs   = smem;   // rows 0..33 <-> tokens t0-1 .. t0+32
    float* tile = smem;   // rows 0..31 (aliased; used after K loop + barrier)

    const int tid  = threadIdx.x;
    const int bIdx = blockIdx.x / (T_ / MROWS);
    const int t0   = (blockIdx.x % (T_ / MROWS)) * MROWS;

    // --- Stage interior rows (tokens t0..t0+31, always in-range) with the
    //     CDNA5 async global->LDS mover: 3072 x b128, 12 per thread. ---
    {
        const float* gbase = in + ((size_t)bIdx * T_ + t0) * D_;
        #pragma unroll
        for (int j = 0; j < (MROWS * D_ / 4) / 256; ++j) {
            const int c = tid + j * 256;                  // float4 chunk id
            const uint32_t ldst = lds_addr_of(&xs[D_ + 4 * c]);
            const float*   gsrc = gbase + 4 * c;
            asm volatile("global_load_async_to_lds_b128 %0, %1, off"
                         :: "v"(ldst), "v"(gsrc) : "memory");
        }
    }
    // --- Halo rows 0 (token t0-1) and 33 (token t0+32): guarded path. ---
    for (int i = tid; i < 2 * D_; i += 256) {
        const int rr  = (i < D_) ? 0 : (MROWS + 1);
        const int c   = (i < D_) ? i : (i - D_);
        const int tok = t0 + rr - 1;
        float v = 0.0f;
        if (tok >= 0 && tok < T_)
            v = in[((size_t)bIdx * T_ + tok) * D_ + c];
        xs[rr * D_ + c] = v;
    }
    asm volatile("s_wait_asynccnt 0x0" ::: "memory");
    __syncthreads();

    const int lane = tid & 31;
    const int wave = tid >> 5;
    const int half = lane >> 4;       // 0: lanes 0-15, 1: lanes 16-31
    const int lm   = lane & 15;
    const int n0   = wave * 48;       // wave owns output cols n0 .. n0+47

    v8f acc00 = {}, acc01 = {}, acc02 = {};   // rows 0-15
    v8f acc10 = {}, acc11 = {}, acc12 = {};   // rows 16-31

    for (int kk0 = 0; kk0 < KDIM; kk0 += 4) {
        const int tap = (kk0 >= 2 * D_) ? 2 : (kk0 >= D_ ? 1 : 0);
        const int ci0 = kk0 - tap * D_;

        // A fragments for both row-tiles (ds_load_b64 each).
        const v2f a0 = *(const v2f*)&xs[(lm + tap) * D_ + ci0 + 2 * half];
        const v2f a1 = *(const v2f*)&xs[(lm + 16 + tap) * D_ + ci0 + 2 * half];

        // B fragments from global (L2-resident); each feeds two WMMAs.
        const float* wrow = W + (size_t)(tap * D_ + ci0 + 2 * half) * D_;
        {
            const int col = n0 + lm;
            v2f b; b.x = wrow[col]; b.y = wrow[D_ + col];
            acc00 = __builtin_amdgcn_wmma_f32_16x16x4_f32(
                false, a0, false, b, (short)0, acc00, false, false);
            acc10 = __builtin_amdgcn_wmma_f32_16x16x4_f32(
                false, a1, false, b, (short)0, acc10, false, false);
        }
        {
            const int col = n0 + 16 + lm;
            v2f b; b.x = wrow[col]; b.y = wrow[D_ + col];
            acc01 = __builtin_amdgcn_wmma_f32_16x16x4_f32(
                false, a0, false, b, (short)0, acc01, false, false);
            acc11 = __builtin_amdgcn_wmma_f32_16x16x4_f32(
                false, a1, false, b, (short)0, acc11, false, false);
        }
        {
            const int col = n0 + 32 + lm;
            v2f b; b.x = wrow[col]; b.y = wrow[D_ + col];
            acc02 = __builtin_amdgcn_wmma_f32_16x16x4_f32(
                false, a0, false, b, (short)0, acc02, false, false);
            acc12 = __builtin_amdgcn_wmma_f32_16x16x4_f32(
                false, a1, false, b, (short)0, acc12, false, false);
        }
    }

    __syncthreads();   // slab reads done everywhere before tile overwrites it

    // Bias + ReLU, scatter accumulators into the LDS tile (C/D layout).
    {
        const int c0 = n0 + lm, c1 = n0 + 16 + lm, c2 = n0 + 32 + lm;
        const float b0 = bias[c0], b1 = bias[c1], b2 = bias[c2];
        #pragma unroll
        for (int j = 0; j < 8; ++j) {
            const int r0 = j + 8 * half;
            const int r1 = 16 + r0;
            float v;
            v = acc00[j] + b0; tile[r0 * D_ + c0] = v > 0.0f ? v : 0.0f;
            v = acc01[j] + b1; tile[r0 * D_ + c1] = v > 0.0f ? v : 0.0f;
            v = acc02[j] + b2; tile[r0 * D_ + c2] = v > 0.0f ? v : 0.0f;
            v = acc10[j] + b0; tile[r1 * D_ + c0] = v > 0.0f ? v : 0.0f;
            v = acc11[j] + b1; tile[r1 * D_ + c1] = v > 0.0f ? v : 0.0f;
            v = acc12[j] + b2; tile[r1 * D_ + c2] = v > 0.0f ? v : 0.0f;
        }
    }
    __syncthreads();

    // LayerNorm over the 384 channels per row; 8 threads per row, 32 rows.
    const int row = tid >> 3;
    const int sub = tid & 7;
    {
        float s1 = 0.0f, s2 = 0.0f;
        for (int c = sub; c < D_; c += 8) {
            const float v = tile[row * D_ + c];
            s1 += v;
            s2 += v * v;
        }
        partS[row * 8 + sub] = s1;
        partQ[row * 8 + sub] = s2;
    }
    __syncthreads();
    if (sub == 0) {
        float s1 = 0.0f, s2 = 0.0f;
        #pragma unroll
        for (int i = 0; i < 8; ++i) {
            s1 += partS[row * 8 + i];
            s2 += partQ[row * 8 + i];
        }
        const float m = s1 * (1.0f / D_);
        const float v = s2 * (1.0f / D_) - m * m;   // population var (jnp.var)
        rowMean[row] = m;
        rowRstd[row] = rsqrtf(v + EPS_);
    }
    __syncthreads();

    const float m = rowMean[row];
    const float r = rowRstd[row];
    float* orow = out + ((size_t)bIdx * T_ + t0 + row) * D_;
    for (int c = sub; c < D_; c += 8)
        orow[c] = (tile[row * D_ + c] - m) * r * gam[c] + bet[c];
}

// ---------------------------------------------------------------------------
// 384->1 projection, log_pred out, durations, per-batch cumsum.
// rintf == round-half-to-even (matches jnp.round).
// ---------------------------------------------------------------------------
__global__ __launch_bounds__(256) void dur_kernel(
    const float* __restrict__ h2,    // [B, T, D]
    const float* __restrict__ lw,    // [D]
    const float* __restrict__ lb,    // [1]
    float* __restrict__ logp_out,    // [B, T]
    int* __restrict__ cum)           // [B, T]
{
    __shared__ float lws[D_];
    __shared__ int   durS[T_];

    const int b   = blockIdx.x;
    const int tid = threadIdx.x;

    for (int i = tid; i < D_; i += 256) lws[i] = lw[i];
    __syncthreads();

    const float lbv = lb[0];
    for (int t = tid; t < T_; t += 256) {
        const float* row = h2 + ((size_t)b * T_ + t) * D_;
        float s = 0.0f;
        for (int c = 0; c < D_; ++c) s += row[c] * lws[c];
        s += lbv;
        logp_out[b * T_ + t] = s;
        durS[t] = (int)rintf(expf(s));
    }
    __syncthreads();

    if (tid == 0) {
        int run = 0;
        for (int t = 0; t < T_; ++t) { run += durS[t]; durS[t] = run; }
    }
    __syncthreads();

    for (int t = tid; t < T_; t += 256) cum[b * T_ + t] = durS[t];
}

// ---------------------------------------------------------------------------
// Length regulation: one wave per output frame; binary search
// (searchsorted-right) on the LDS-cached cumsum, then stream the 1536-byte
// row as float4 (global b128) with the tail mask. 226 MB HBM-bound stage.
// ---------------------------------------------------------------------------
__global__ __launch_bounds__(256) void regulate_kernel(
    const float* __restrict__ x,    // [B, T, D]
    const int* __restrict__ cum,    // [B, T]
    float* __restrict__ out)        // [B, MAXF, D]
{
    __shared__ int cs[T_];

    const int b   = blockIdx.y;
    const int tid = threadIdx.x;

    for (int i = tid; i < T_; i += 256) cs[i] = cum[b * T_ + i];
    __syncthreads();

    const int wave = tid >> 5;
    const int lane = tid & 31;
    const int f    = blockIdx.x * 8 + wave;

    // first idx with cs[idx] > f  (jnp.searchsorted side='right')
    int lo = 0, hi = T_;
    while (lo < hi) {
        const int mid = (lo + hi) >> 1;
        if (cs[mid] <= f) lo = mid + 1; else hi = mid;
    }
    const int   idx  = lo < (T_ - 1) ? lo : (T_ - 1);
    const float mask = (f < cs[T_ - 1]) ? 1.0f : 0.0f;

    const float4* src = (const float4*)(x + ((size_t)b * T_ + idx) * D_);
    float4*       dst = (float4*)(out + ((size_t)b * MAXF + f) * D_);
    #pragma unroll
    for (int j = 0; j < D_ / 4 / 32; ++j) {
        float4 v = src[lane + j * 32];
        v.x *= mask; v.y *= mask; v.z *= mask; v.w *= mask;
        dst[lane + j * 32] = v;
    }
}

// ---------------------------------------------------------------------------
// Inputs (setup_inputs order):
//  0 x [B,T,D]  1 conv1_w [3,D,D]  2 conv1_b [D]  3 ln1_g [D]  4 ln1_b [D]
//  5 conv2_w    6 conv2_b          7 ln2_g        8 ln2_b
//  9 lin_w [D]  10 lin_b [1]
// d_out = out [B,MAXF,D] fp32  ++  log_pred [B,T] fp32
// d_ws  = h1 (50.3MB) ++ h2 (50.3MB) ++ cum (128KB int32)
// ---------------------------------------------------------------------------
extern "C" void kernel_launch(void* const* d_in, const int* in_sizes, int n_in,
                              void* d_out, int out_size, void* d_ws, size_t ws_size,
                              hipStream_t stream) {
    const float* x   = (const float*)d_in[0];
    const float* c1w = (const float*)d_in[1];
    const float* c1b = (const float*)d_in[2];
    const float* l1g = (const float*)d_in[3];
    const float* l1b = (const float*)d_in[4];
    const float* c2w = (const float*)d_in[5];
    const float* c2b = (const float*)d_in[6];
    const float* l2g = (const float*)d_in[7];
    const float* l2b = (const float*)d_in[8];
    const float* lw  = (const float*)d_in[9];
    const float* lb  = (const float*)d_in[10];

    const size_t NH = (size_t)B_ * T_ * D_;
    float* ws  = (float*)d_ws;
    float* h1  = ws;
    float* h2  = ws + NH;
    int*   cum = (int*)(ws + 2 * NH);

    float* outp = (float*)d_out;
    float* logp = outp + (size_t)B_ * MAXF * D_;

    const int convBlocks = B_ * T_ / MROWS;   // 1024
    conv_ln_kernel<<<convBlocks, 256, 0, stream>>>(x,  c1w, c1b, l1g, l1b, h1);
    conv_ln_kernel<<<convBlocks, 256, 0, stream>>>(h1, c2w, c2b, l2g, l2b, h2);
    dur_kernel<<<B_, 256, 0, stream>>>(h2, lw, lb, logp, cum);
    regulate_kernel<<<dim3(MAXF / 8, B_), 256, 0, stream>>>(x, cum, outp);
}